// GraphSAGE_38216619000106
// MI455X (gfx1250) — compile-verified
//
#include <hip/hip_runtime.h>
#include <hip/hip_bf16.h>

// ---------------------------------------------------------------------------
// GraphSAGE 2-layer forward for MI455X (gfx1250, wave32)
//   layer: out = mean_agg(x) @ W_l^T + b + x @ W_r^T    (+ReLU for layer 1)
//   final: h2  and log_softmax(h2)
//
// Strategy (from roofline): GEMMs are ~10 GFLOP total -> irrelevant; use exact
// fp32 WMMA (V_WMMA_F32_16X16X4_F32) so semantics match the f32 reference.
// The dominant cost is the edge gather/scatter (~2.5 GB of L2 traffic); node
// feature tables (25.6/51.2 MB) live in the 192 MB L2, so a coalesced
// one-wave-per-edge atomicAdd scatter is bandwidth-optimal.
// ---------------------------------------------------------------------------

#define N_NODES 50000
#define N_EDGES 800000
#define D_IN    128
#define D_H     256
#define D_OUT   64

typedef __attribute__((ext_vector_type(2))) float v2f;
typedef __attribute__((ext_vector_type(8))) float v8f;

// ---------------------------------------------------------------- zero fill
__global__ void zero_f32(float* __restrict__ p, long n) {
    long i = (long)blockIdx.x * blockDim.x + threadIdx.x;
    long stride = (long)gridDim.x * blockDim.x;
    for (; i < n; i += stride) p[i] = 0.0f;
}

// ---------------------------------------------------------------- degrees
__global__ void degree_kernel(const int* __restrict__ dst, float* __restrict__ deg) {
    int e = blockIdx.x * blockDim.x + threadIdx.x;
    if (e < N_EDGES) atomicAdd(&deg[dst[e]], 1.0f);
}

// ------------------------------------------------- edge scatter-sum (gather+atomicAdd)
// One wave32 per edge; lane strides give fully coalesced 128B bursts per step.
template <int D>
__global__ void scatter_sum_kernel(const float* __restrict__ x,
                                   const int* __restrict__ src,
                                   const int* __restrict__ dst,
                                   float* __restrict__ out) {
    int e = blockIdx.x * (blockDim.x >> 5) + (threadIdx.x >> 5);
    if (e >= N_EDGES) return;
    int lane = threadIdx.x & 31;
    int s = src[e];
    int d = dst[e];
    const float* __restrict__ xs = x + (long)s * D;
    float* __restrict__ od = out + (long)d * D;
#pragma unroll
    for (int v = 0; v < D / 32; ++v) {
        atomicAdd(&od[lane + v * 32], xs[lane + v * 32]);
    }
}

// ------------------------------------------------- fused SAGE GEMM via fp32 WMMA
// out[i,j] = relu?( (sum[i,:]*rdeg[i]) . W_l[j,:] + b[j] + xin[i,:] . W_r[j,:] )
//
// One wave computes one 16x16 C tile with V_WMMA_F32_16X16X4_F32.
// fp32 A-frag layout (ISA 7.12.2):  lane half = lane>>4, r = lane&15
//   a[0] = A[M=r][k0 + 2*half],  a[1] = A[M=r][k0 + 2*half + 1]
// B[k][n] = W[n][k], so the B-frag is the same consecutive-k pair read from W.
// fp32 C/D layout: VGPR v -> C[M = v + 8*half][N = r].
template <int K, int NOUT, bool RELU>
__global__ __launch_bounds__(128) void sage_gemm_kernel(
    const float* __restrict__ xin,   // [N_NODES, K]   root features
    const float* __restrict__ sum,   // [N_NODES, K]   scatter-summed neighbors
    const float* __restrict__ deg,   // [N_NODES]
    const float* __restrict__ W_l,   // [NOUT, K]
    const float* __restrict__ W_r,   // [NOUT, K]
    const float* __restrict__ bias,  // [NOUT]
    float* __restrict__ out)         // [N_NODES, NOUT]
{
    const int wave = threadIdx.x >> 5;            // 4 waves / block
    const int lane = threadIdx.x & 31;
    const int mt   = blockIdx.x;                  // 0..3124 (50000/16 exact)
    const int nt   = blockIdx.y * 4 + wave;       // N tile
    const int half = lane >> 4;
    const int r    = lane & 15;

    const int row = mt * 16 + r;                  // A row for this lane
    const int col = nt * 16 + r;                  // B col (== C col) for this lane

    const float dg   = deg[row];
    const float rdeg = 1.0f / fmaxf(dg, 1.0f);

    v8f c;
    {
        const float bv = bias[col];
#pragma unroll
        for (int i = 0; i < 8; ++i) c[i] = bv;
    }

    // ---- pass 1: mean-aggregated neighbors x W_l^T --------------------------
    {
        const float* __restrict__ Arow = sum + (long)row * K;
        const float* __restrict__ Brow = W_l + (long)col * K;
#pragma unroll 4
        for (int k0 = 0; k0 < K; k0 += 4) {
            v2f a, b;
            a[0] = Arow[k0 + 2 * half + 0] * rdeg;
            a[1] = Arow[k0 + 2 * half + 1] * rdeg;
            b[0] = Brow[k0 + 2 * half + 0];
            b[1] = Brow[k0 + 2 * half + 1];
            c = __builtin_amdgcn_wmma_f32_16x16x4_f32(
                /*neg_a=*/false, a, /*neg_b=*/false, b,
                /*c_mod=*/(short)0, c, /*reuse_a=*/false, /*reuse_b=*/false);
        }
    }
    // ---- pass 2: root features x W_r^T (same accumulator) -------------------
    {
        const float* __restrict__ Arow = xin + (long)row * K;
        const float* __restrict__ Brow = W_r + (long)col * K;
#pragma unroll 4
        for (int k0 = 0; k0 < K; k0 += 4) {
            v2f a, b;
            a[0] = Arow[k0 + 2 * half + 0];
            a[1] = Arow[k0 + 2 * half + 1];
            b[0] = Brow[k0 + 2 * half + 0];
            b[1] = Brow[k0 + 2 * half + 1];
            c = __builtin_amdgcn_wmma_f32_16x16x4_f32(
                false, a, false, b, (short)0, c, false, false);
        }
    }

    // ---- store D tile -------------------------------------------------------
#pragma unroll
    for (int v = 0; v < 8; ++v) {
        const int m = mt * 16 + v + 8 * half;
        float val = c[v];
        if (RELU) val = fmaxf(val, 0.0f);
        out[(long)m * NOUT + col] = val;
    }
}

// ------------------------------------------------- log_softmax over 64 cols
// One wave32 per node; 2 values per lane; shfl-xor tree reductions.
__global__ void logsoftmax_kernel(const float* __restrict__ h,
                                  float* __restrict__ out) {
    int node = blockIdx.x * (blockDim.x >> 5) + (threadIdx.x >> 5);
    if (node >= N_NODES) return;
    int lane = threadIdx.x & 31;
    const float* __restrict__ row = h + (long)node * D_OUT;
    float v0 = row[lane];
    float v1 = row[lane + 32];

    float m = fmaxf(v0, v1);
#pragma unroll
    for (int off = 16; off > 0; off >>= 1)
        m = fmaxf(m, __shfl_xor(m, off, 32));

    float s = __expf(v0 - m) + __expf(v1 - m);
#pragma unroll
    for (int off = 16; off > 0; off >>= 1)
        s += __shfl_xor(s, off, 32);

    float lse = m + __logf(s);
    out[(long)node * D_OUT + lane]      = v0 - lse;
    out[(long)node * D_OUT + lane + 32] = v1 - lse;
}

// ---------------------------------------------------------------------------
extern "C" void kernel_launch(void* const* d_in, const int* in_sizes, int n_in,
                              void* d_out, int out_size, void* d_ws, size_t ws_size,
                              hipStream_t stream) {
    const float* x    = (const float*)d_in[0];            // [50000,128]
    const int*   ei   = (const int*)d_in[1];              // [2,800000]
    const float* W1_l = (const float*)d_in[2];            // [256,128]
    const float* W1_r = (const float*)d_in[3];            // [256,128]
    const float* b1   = (const float*)d_in[4];            // [256]
    const float* W2_l = (const float*)d_in[5];            // [64,256]
    const float* W2_r = (const float*)d_in[6];            // [64,256]
    const float* b2   = (const float*)d_in[7];            // [64]

    const int* src = ei;
    const int* dst = ei + N_EDGES;

    // workspace carve (~128.2 MB)
    float* deg  = (float*)d_ws;                           // [50000]
    float* sum1 = deg  + N_NODES;                         // [50000,128]
    float* h1   = sum1 + (size_t)N_NODES * D_IN;          // [50000,256]
    float* sum2 = h1   + (size_t)N_NODES * D_H;           // [50000,256]

    float* h2     = (float*)d_out;                        // [50000,64]
    float* logsm  = h2 + (size_t)N_NODES * D_OUT;         // [50000,64]

    // ---- zero accumulators (deg+sum1 are contiguous) ----
    zero_f32<<<2048, 256, 0, stream>>>(deg, (long)N_NODES + (long)N_NODES * D_IN);
    zero_f32<<<2048, 256, 0, stream>>>(sum2, (long)N_NODES * D_H);

    // ---- degrees ----
    degree_kernel<<<(N_EDGES + 255) / 256, 256, 0, stream>>>(dst, deg);

    // ---- layer 1: scatter + fused WMMA GEMM + ReLU ----
    scatter_sum_kernel<D_IN><<<(N_EDGES + 7) / 8, 256, 0, stream>>>(x, src, dst, sum1);
    {
        dim3 grid(N_NODES / 16, (D_H / 16) / 4);          // (3125, 4)
        sage_gemm_kernel<D_IN, D_H, true><<<grid, 128, 0, stream>>>(
            x, sum1, deg, W1_l, W1_r, b1, h1);
    }

    // ---- layer 2: scatter + fused WMMA GEMM ----
    scatter_sum_kernel<D_H><<<(N_EDGES + 7) / 8, 256, 0, stream>>>(h1, src, dst, sum2);
    {
        dim3 grid(N_NODES / 16, (D_OUT / 16) / 4);        // (3125, 1)
        sage_gemm_kernel<D_H, D_OUT, false><<<grid, 128, 0, stream>>>(
            h1, sum2, deg, W2_l, W2_r, b2, h2);
    }

    // ---- log_softmax ----
    logsoftmax_kernel<<<(N_NODES + 7) / 8, 256, 0, stream>>>(h2, logsm);
}